// SEDR_module_32813550141702
// MI455X (gfx1250) — compile-verified
//
#include <hip/hip_runtime.h>
#include <hip/hip_bf16.h>

typedef __attribute__((ext_vector_type(2))) float v2f;
typedef __attribute__((ext_vector_type(8))) float v8f;
typedef __attribute__((ext_vector_type(4))) unsigned int v4u;
typedef __attribute__((ext_vector_type(8))) int v8i;
typedef __attribute__((ext_vector_type(4))) int v4i;

#define BN_EPS 1e-3f
#define TDM_KC 64   // K-rows of B staged per TDM chunk (64 x 64 f32 = 16KB LDS)

// ---------------------------------------------------------------------------
// Utility kernels
// ---------------------------------------------------------------------------
__global__ void zero_f32(float* __restrict__ p, long n) {
  long i = (long)blockIdx.x * blockDim.x + threadIdx.x;
  if (i < n) p[i] = 0.0f;
}

__global__ void scatter_flags(const int* __restrict__ idx, float* __restrict__ flags, int m) {
  int i = blockIdx.x * blockDim.x + threadIdx.x;
  if (i < m) flags[idx[i]] = 1.0f;
}

// ---------------------------------------------------------------------------
// Shared epilogue: EPI 0 = identity, 1 = Linear-bias + BatchNorm(eval) + ELU,
// 2 = ReLU.  C/D layout: lane L holds column L%16, rows r + 8*(L/16).
// ---------------------------------------------------------------------------
template <int EPI>
__device__ __forceinline__ void store_tile(v8f acc, float* __restrict__ Out,
                                           int row0, int lane, int ncol, int NB,
                                           const float* bias, const float* gamma,
                                           const float* beta, const float* rmean,
                                           const float* rvar) {
  float sc = 1.0f, sh = 0.0f;
  if (EPI == 1) {
    float s = gamma[ncol] * rsqrtf(rvar[ncol] + BN_EPS);
    sc = s;
    sh = (bias[ncol] - rmean[ncol]) * s + beta[ncol];
  }
  const int rbase = row0 + ((lane >> 4) << 3);
#pragma unroll
  for (int r = 0; r < 8; ++r) {
    float v = acc[r];
    if (EPI == 1) { v = v * sc + sh; v = (v > 0.0f) ? v : (__expf(v) - 1.0f); }
    else if (EPI == 2) { v = fmaxf(v, 0.0f); }
    Out[(long)(rbase + r) * NB + ncol] = v;
  }
}

// ---------------------------------------------------------------------------
// TDM-staged fp32 WMMA GEMM (for the large-K layer): B chunks of TDM_KC rows
// are DMA'd into LDS by the Tensor Data Mover (tensor_load_to_lds, waited with
// s_wait_tensorcnt), then all waves consume B fragments via ds_load while
// streaming A rows from global (with unconditional speculative prefetch).
// One wave computes one 16x16 tile with V_WMMA_F32_16X16X4_F32.
// Requires NB % 16 == 0, K % 4 == 0.
// ---------------------------------------------------------------------------
template <int EPI, bool MASK_ADD>
__global__ __launch_bounds__(256)
void wmma_gemm_f32_tdm(const float* __restrict__ A, const float* __restrict__ B,
                       float* __restrict__ Out, int M, int K, int NB,
                       const float* __restrict__ flags, const float* __restrict__ token,
                       const float* __restrict__ bias,  const float* __restrict__ gamma,
                       const float* __restrict__ beta,  const float* __restrict__ rmean,
                       const float* __restrict__ rvar) {
  extern __shared__ float sB[];                 // TDM_KC * NB floats
  const int lane   = threadIdx.x & 31;
  const int wave   = threadIdx.x >> 5;
  const int tilesN = NB >> 4;
  int tile = blockIdx.x * (blockDim.x >> 5) + wave;
  const bool active = tile < (M >> 4) * tilesN;
  if (!active) tile = 0;                        // keep loads valid; no store later

  const int mt   = tile / tilesN;
  const int nt   = tile - mt * tilesN;
  const int row0 = mt << 4;
  const int mrow = row0 + (lane & 15);
  const int kh   = (lane >> 4) << 1;            // 0 or 2: K-pair this lane holds
  const int ncol = (nt << 4) + (lane & 15);

  const float flagv = MASK_ADD ? flags[mrow] : 0.0f;
  const float* __restrict__ arow = A + (long)mrow * K + kh;
  const float* __restrict__ trow = MASK_ADD ? (token + kh) : nullptr;
  const unsigned int lds_base = (unsigned int)(unsigned long long)(uintptr_t)sB;

  v8f acc = {0.f, 0.f, 0.f, 0.f, 0.f, 0.f, 0.f, 0.f};
  for (int kc = 0; kc < K; kc += TDM_KC) {
    const int kcLen = (K - kc < TDM_KC) ? (K - kc) : TDM_KC;
    __syncthreads();                            // previous chunk fully consumed
    if (wave == 0) {
      // Tensor DMA descriptor (D#), ISA ch.8: 2D tensor = kcLen x NB f32 tile.
      unsigned long long ga = (unsigned long long)(uintptr_t)(B + (long)kc * NB);
      v4u g0;
      g0[0] = 1u;                                               // count=1
      g0[1] = lds_base;                                         // lds_addr
      g0[2] = (unsigned int)ga;                                 // global_addr lo
      g0[3] = (unsigned int)((ga >> 32) & 0x1FFFFFFull) | 0x80000000u; // hi|type=2
      v8i g1;
      g1[0] = 0x00020000;                                       // data_size=4B
      g1[1] = (NB & 0xFFFF) << 16;                              // tensor_dim0 lo
      g1[2] = ((unsigned)NB >> 16) | ((kcLen & 0xFFFF) << 16);  // dim0 hi | dim1 lo
      g1[3] = (NB & 0xFFFF) << 16;                              // dim1 hi | tile_dim0
      g1[4] = kcLen & 0xFFFF;                                   // tile_dim1 (tile_dim2=0)
      g1[5] = NB;                                               // dim0_stride lo32
      g1[6] = 0;
      g1[7] = 0;
      v4i g2 = {0, 0, 0, 0}, g3 = {0, 0, 0, 0};
      v8i g4 = {0, 0, 0, 0, 0, 0, 0, 0};
      __builtin_amdgcn_tensor_load_to_lds(g0, g1, g2, g3, g4, 0); // TENSORcnt++
      __builtin_amdgcn_s_wait_tensorcnt(0);                       // chunk landed
    }
    __syncthreads();                            // chunk visible to all waves

    const float* lB = sB + kh * NB + ncol;
    for (int k = 0; k < kcLen; k += 4) {
      v2f a, b;
      a[0] = arow[0];
      a[1] = arow[1];
      if (MASK_ADD) {
        a[0] = fmaf(flagv, trow[0], a[0]);
        a[1] = fmaf(flagv, trow[1], a[1]);
        trow += 4;
      }
      b[0] = lB[0];
      b[1] = lB[NB];
      __builtin_prefetch(arow + 64, 0, 3);      // speculative A-stream prefetch
      acc = __builtin_amdgcn_wmma_f32_16x16x4_f32(
          false, a, false, b, (short)0, acc, false, false);
      arow += 4;
      lB += 4 * NB;
    }
  }
  if (!active) return;
  store_tile<EPI>(acc, Out, row0, lane, ncol, NB, bias, gamma, beta, rmean, rvar);
}

// ---------------------------------------------------------------------------
// Generic fp32 WMMA GEMM: Out[M,NB] = epilogue(A[M,K] @ B[K,NB])
// TAIL:    NB % 16 != 0 -> branchless clamp + zero-scale on B, guarded store.
// USE_LDS: stage the whole B matrix in LDS cooperatively (K*NB*4 <= 64KB).
// ---------------------------------------------------------------------------
template <int EPI, bool TAIL, bool USE_LDS>
__global__ __launch_bounds__(256)
void wmma_gemm_f32(const float* __restrict__ A, const float* __restrict__ B,
                   float* __restrict__ Out, int M, int K, int NB,
                   const float* __restrict__ bias,  const float* __restrict__ gamma,
                   const float* __restrict__ beta,  const float* __restrict__ rmean,
                   const float* __restrict__ rvar) {
  extern __shared__ float sB[];
  if (USE_LDS) {
    for (int i = threadIdx.x; i < K * NB; i += blockDim.x) sB[i] = B[i];
    __syncthreads();
  }
  const int lane   = threadIdx.x & 31;
  const int wave   = threadIdx.x >> 5;
  const int tilesN = (NB + 15) >> 4;
  const int tile   = blockIdx.x * (blockDim.x >> 5) + wave;
  if (tile >= (M >> 4) * tilesN) return;        // wave-uniform

  const int mt   = tile / tilesN;
  const int nt   = tile - mt * tilesN;
  const int row0 = mt << 4;
  const int mrow = row0 + (lane & 15);
  const int kh   = (lane >> 4) << 1;
  int   ncol   = (nt << 4) + (lane & 15);
  bool  colOK  = true;
  float bscale = 1.0f;
  if (TAIL && ncol >= NB) { colOK = false; bscale = 0.0f; ncol = NB - 1; }

  const float* __restrict__ arow = A + (long)mrow * K + kh;
  const float* __restrict__ gB   = B + (long)kh * NB + ncol;
  const float*              lB   = sB + kh * NB + ncol;

  v8f acc = {0.f, 0.f, 0.f, 0.f, 0.f, 0.f, 0.f, 0.f};
  for (int k = 0; k < K; k += 4) {
    v2f a, b;
    a[0] = arow[0];
    a[1] = arow[1];
    if (USE_LDS) { b[0] = lB[0]; b[1] = lB[NB]; lB += 4 * NB; }
    else         { b[0] = gB[0]; b[1] = gB[NB]; gB += 4 * NB; }
    if (TAIL) { b[0] *= bscale; b[1] *= bscale; }   // branchless tail masking
    acc = __builtin_amdgcn_wmma_f32_16x16x4_f32(
        false, a, false, b, (short)0, acc, false, false);
    arow += 4;
  }
  if (TAIL && !colOK) return;                   // store-side guard only
  store_tile<EPI>(acc, Out, row0, lane, ncol, NB, bias, gamma, beta, rmean, rvar);
}

// ---------------------------------------------------------------------------
// Sparse adj @ dense: out[rows[e], :] += vals[e] * dense[cols[e], :]
// ---------------------------------------------------------------------------
__global__ void spmm_atomic(const int* __restrict__ rows, const int* __restrict__ cols,
                            const float* __restrict__ vals, const float* __restrict__ dense,
                            float* __restrict__ out, int F, long total) {
  long idx = (long)blockIdx.x * blockDim.x + threadIdx.x;
  if (idx >= total) return;
  int e = (int)(idx / F);
  int c = (int)(idx - (long)e * F);
  atomicAdd(&out[(long)rows[e] * F + c], vals[e] * dense[(long)cols[e] * F + c]);
}

// z = [feat_x | mu], gnn_z = mu
__global__ void concat_z(const float* __restrict__ fx, const float* __restrict__ mu,
                         float* __restrict__ z, float* __restrict__ gz,
                         int N, int F1, int F2) {
  long idx = (long)blockIdx.x * blockDim.x + threadIdx.x;
  int ZF = F1 + F2;
  if (idx >= (long)N * ZF) return;
  int n = (int)(idx / ZF);
  int j = (int)(idx - (long)n * ZF);
  float v = (j < F1) ? fx[(long)n * F1 + j] : mu[(long)n * F2 + (j - F1)];
  z[idx] = v;
  if (j >= F1) gz[(long)n * F2 + (j - F1)] = v;
}

// Student-t soft assignments (alpha = 1 -> q = 1/(1+d2), normalized)
__global__ void q_kernel(const float* __restrict__ z, const float* __restrict__ cl,
                         float* __restrict__ q, int N, int ZF, int K) {
  extern __shared__ float scl[];
  for (int i = threadIdx.x; i < K * ZF; i += blockDim.x) scl[i] = cl[i];
  __syncthreads();
  int n = blockIdx.x * blockDim.x + threadIdx.x;
  if (n >= N) return;
  float zr[32];
  for (int j = 0; j < ZF; ++j) zr[j] = z[(long)n * ZF + j];
  float qs[16];
  float ssum = 0.0f;
  for (int k = 0; k < K; ++k) {
    float d2 = 0.0f;
    for (int j = 0; j < ZF; ++j) { float d = zr[j] - scl[k * ZF + j]; d2 += d * d; }
    float qv = 1.0f / (1.0f + d2);
    qs[k] = qv; ssum += qv;
  }
  float inv = 1.0f / ssum;
  for (int k = 0; k < K; ++k) q[(long)n * K + k] = qs[k] * inv;
}

// Per-masked-node SCE term: (1 - cos(l2n(x_rec), l2n(x_init)))^3
__global__ __launch_bounds__(256)
void loss_node_kernel(const int* __restrict__ mask_nodes, const float* __restrict__ x,
                      const float* __restrict__ token, const float* __restrict__ de_feat,
                      float* __restrict__ lossvals, int D) {
  __shared__ float sdot[256], snr[256], sni[256];
  const int node = mask_nodes[blockIdx.x];
  const float* __restrict__ xr = de_feat + (long)node * D;
  const float* __restrict__ xi = x + (long)node * D;
  float dot = 0.f, nr = 0.f, ni = 0.f;
  for (int c = threadIdx.x; c < D; c += blockDim.x) {
    float r = xr[c];
    float i0 = xi[c] + token[c];              // out_x on masked rows
    dot += r * i0; nr += r * r; ni += i0 * i0;
  }
  sdot[threadIdx.x] = dot; snr[threadIdx.x] = nr; sni[threadIdx.x] = ni;
  __syncthreads();
  for (int s = blockDim.x >> 1; s > 0; s >>= 1) {
    if (threadIdx.x < s) {
      sdot[threadIdx.x] += sdot[threadIdx.x + s];
      snr[threadIdx.x]  += snr[threadIdx.x + s];
      sni[threadIdx.x]  += sni[threadIdx.x + s];
    }
    __syncthreads();
  }
  if (threadIdx.x == 0) {
    float cosv = sdot[0] / (fmaxf(sqrtf(snr[0]), 1e-12f) * fmaxf(sqrtf(sni[0]), 1e-12f));
    float t = 1.0f - cosv;
    lossvals[blockIdx.x] = t * t * t;
  }
}

// Deterministic final mean (fixed single-block traversal order)
__global__ void loss_reduce(const float* __restrict__ lossvals, float* __restrict__ out, int m) {
  __shared__ float s[256];
  float a = 0.0f;
  for (int i = threadIdx.x; i < m; i += 256) a += lossvals[i];
  s[threadIdx.x] = a;
  __syncthreads();
  for (int st = 128; st > 0; st >>= 1) {
    if (threadIdx.x < st) s[threadIdx.x] += s[threadIdx.x + st];
    __syncthreads();
  }
  if (threadIdx.x == 0) out[0] = s[0] / (float)m;
}

// ---------------------------------------------------------------------------
// Host-side orchestration
// ---------------------------------------------------------------------------
static inline int cdiv(long a, long b) { return (int)((a + b - 1) / b); }

extern "C" void kernel_launch(void* const* d_in, const int* in_sizes, int n_in,
                              void* d_out, int out_size, void* d_ws, size_t ws_size,
                              hipStream_t stream) {
  const float* x      = (const float*)d_in[0];
  const int*   erows  = (const int*)  d_in[1];
  const int*   ecols  = (const int*)  d_in[2];
  const float* evals  = (const float*)d_in[3];
  const int*   masknd = (const int*)  d_in[4];
  const float* token  = (const float*)d_in[5];
  const float* W1 = (const float*)d_in[6];  const float* b1 = (const float*)d_in[7];
  const float* g1 = (const float*)d_in[8];  const float* be1 = (const float*)d_in[9];
  const float* rm1 = (const float*)d_in[10]; const float* rv1 = (const float*)d_in[11];
  const float* W2 = (const float*)d_in[12]; const float* b2 = (const float*)d_in[13];
  const float* g2 = (const float*)d_in[14]; const float* be2 = (const float*)d_in[15];
  const float* rm2 = (const float*)d_in[16]; const float* rv2 = (const float*)d_in[17];
  const float* Wgc1 = (const float*)d_in[18];
  const float* Wgc2 = (const float*)d_in[19];
  const float* Wgc3 = (const float*)d_in[20];
  const float* Wdec = (const float*)d_in[21];
  const float* clus = (const float*)d_in[22];

  const int E    = in_sizes[1];
  const int MSKN = in_sizes[4];
  const int D    = in_sizes[5];
  const int FH1  = in_sizes[7];
  const int FH2  = in_sizes[13];
  const int N    = in_sizes[0] / D;
  const int GH1  = in_sizes[18] / FH2;
  const int GH2  = in_sizes[19] / GH1;
  const int ZF   = FH2 + GH2;
  const int Kc   = in_sizes[22] / ZF;

  float* out = (float*)d_out;
  const long mu_off = (long)N * ZF;
  const long lv_off = mu_off + (long)N * GH2;
  const long de_off = lv_off + (long)N * GH2;
  const long q_off  = de_off + (long)N * D;
  const long fx_off = q_off  + (long)N * Kc;
  const long gz_off = fx_off + (long)N * FH2;
  const long ls_off = gz_off + (long)N * GH2;
  (void)out_size; (void)n_in; (void)ws_size;

  // workspace layout (fp32)
  float* wsf = (float*)d_ws;
  long o = 0;
  float* flags   = wsf + o; o += N;
  float* h       = wsf + o; o += (long)N * FH1;
  float* s1      = wsf + o; o += (long)N * FH2;   // spmm(feat_x)
  float* hidden1 = wsf + o; o += (long)N * GH1;
  float* s2      = wsf + o; o += (long)N * GH1;   // spmm(hidden1)
  float* s3      = wsf + o; o += (long)N * ZF;    // spmm(z)
  float* lvals   = wsf + o; o += MSKN;

  float* feat_x = out + fx_off;
  float* mu     = out + mu_off;
  float* lgv    = out + lv_off;
  float* zbuf   = out;                             // z at offset 0
  float* gz     = out + gz_off;
  float* defeat = out + de_off;

  const int TPB = 256, WPB = 8;
  auto gemm_blocks = [&](int M, int NB) { return cdiv((long)(M >> 4) * ((NB + 15) >> 4), WPB); };

  // 0) mask flags
  zero_f32<<<cdiv(N, TPB), TPB, 0, stream>>>(flags, N);
  scatter_flags<<<cdiv(MSKN, TPB), TPB, 0, stream>>>(masknd, flags, MSKN);

  // 1) h = ELU(BN(out_x @ W1 + b1)) — K=1000; B chunks DMA'd to LDS by the TDM
  wmma_gemm_f32_tdm<1, true>
      <<<gemm_blocks(N, FH1), TPB, (size_t)TDM_KC * FH1 * sizeof(float), stream>>>(
      x, W1, h, N, D, FH1, flags, token, b1, g1, be1, rm1, rv1);

  // 2) feat_x = ELU(BN(h @ W2 + b2)) — B staged in LDS (4KB)
  wmma_gemm_f32<1, false, true>
      <<<gemm_blocks(N, FH2), TPB, (size_t)FH1 * FH2 * sizeof(float), stream>>>(
      h, W2, feat_x, N, FH1, FH2, b2, g2, be2, rm2, rv2);

  // 3) hidden1 = relu(spmm(feat_x @ Wgc1)) == relu(spmm(feat_x) @ Wgc1)
  zero_f32<<<cdiv((long)N * FH2, TPB), TPB, 0, stream>>>(s1, (long)N * FH2);
  spmm_atomic<<<cdiv((long)E * FH2, TPB), TPB, 0, stream>>>(
      erows, ecols, evals, feat_x, s1, FH2, (long)E * FH2);
  wmma_gemm_f32<2, false, true>
      <<<gemm_blocks(N, GH1), TPB, (size_t)FH2 * GH1 * sizeof(float), stream>>>(
      s1, Wgc1, hidden1, N, FH2, GH1, nullptr, nullptr, nullptr, nullptr, nullptr);

  // 4) mu / logvar = spmm(hidden1) @ Wgc2 / Wgc3
  zero_f32<<<cdiv((long)N * GH1, TPB), TPB, 0, stream>>>(s2, (long)N * GH1);
  spmm_atomic<<<cdiv((long)E * GH1, TPB), TPB, 0, stream>>>(
      erows, ecols, evals, hidden1, s2, GH1, (long)E * GH1);
  wmma_gemm_f32<0, false, true>
      <<<gemm_blocks(N, GH2), TPB, (size_t)GH1 * GH2 * sizeof(float), stream>>>(
      s2, Wgc2, mu, N, GH1, GH2, nullptr, nullptr, nullptr, nullptr, nullptr);
  wmma_gemm_f32<0, false, true>
      <<<gemm_blocks(N, GH2), TPB, (size_t)GH1 * GH2 * sizeof(float), stream>>>(
      s2, Wgc3, lgv, N, GH1, GH2, nullptr, nullptr, nullptr, nullptr, nullptr);

  // 5) z = [feat_x | mu], gnn_z = mu
  concat_z<<<cdiv((long)N * ZF, TPB), TPB, 0, stream>>>(feat_x, mu, zbuf, gz, N, FH2, GH2);

  // 6) de_feat = spmm(z @ Wdec) == spmm(z) @ Wdec  (avoids 320MB intermediate + 5GB gather)
  zero_f32<<<cdiv((long)N * ZF, TPB), TPB, 0, stream>>>(s3, (long)N * ZF);
  spmm_atomic<<<cdiv((long)E * ZF, TPB), TPB, 0, stream>>>(
      erows, ecols, evals, zbuf, s3, ZF, (long)E * ZF);
  if (D % 16 == 0)
    wmma_gemm_f32<0, false, false><<<gemm_blocks(N, D), TPB, 0, stream>>>(
        s3, Wdec, defeat, N, ZF, D, nullptr, nullptr, nullptr, nullptr, nullptr);
  else
    wmma_gemm_f32<0, true, false><<<gemm_blocks(N, D), TPB, 0, stream>>>(
        s3, Wdec, defeat, N, ZF, D, nullptr, nullptr, nullptr, nullptr, nullptr);

  // 7) q soft assignments
  q_kernel<<<cdiv(N, TPB), TPB, (size_t)Kc * ZF * sizeof(float), stream>>>(
      zbuf, clus, out + q_off, N, ZF, Kc);

  // 8) SCE loss over masked nodes
  loss_node_kernel<<<MSKN, TPB, 0, stream>>>(masknd, x, token, defeat, lvals, D);
  loss_reduce<<<1, 256, 0, stream>>>(lvals, out + ls_off, MSKN);
}